// TransientCombNoise_32573031973082
// MI455X (gfx1250) — compile-verified
//
#include <hip/hip_runtime.h>
#include <stdint.h>

// TransientCombNoise for MI455X (gfx1250).
// Pure-bandwidth workload (~34 MB traffic ~= 1.5us at 23.3 TB/s): the comb
// filter with delay in [33,63] over 64 samples is
//   y[k] = n[k]*exp(-k/atk)*E + (k>=d ? tilt*y[k-d] : 0);  out = y / rms(y)
// Strategy: CDNA5 async-DMA staging (global_load_async_to_lds_b128) for
// coalesced HBM reads, padded LDS rows (pitch 68 floats = 272B: 16B-aligned
// rows for b128 DMA, max 2-way bank conflict on per-row reads) for the
// dynamic delayed tap, hardware v_exp_f32 for the envelope, and
// global_store_async_from_lds_b128 for coalesced write-back.
// Sync: s_wait_asynccnt + workgroup barrier.

#define BLK          64     // samples per row
#define TPB          128    // threads per block (4 waves, wave32)
#define ROWS_PB      128    // one row per thread
#define ROW_PAD      68     // floats of LDS pitch per row (272B, multiple of 16)
#define NROUNDS      16     // (ROWS_PB*BLK) / (TPB*4) float4 transfers per lane

__global__ __launch_bounds__(TPB)
void transient_comb_kernel(const float* __restrict__ params,
                           const float* __restrict__ noise,
                           float* __restrict__ out,
                           int nrows)
{
    __shared__ __align__(16) float smem[ROWS_PB * ROW_PAD];

    const int tid = threadIdx.x;
    const int r0  = blockIdx.x * ROWS_PB;
    const bool full = (r0 + ROWS_PB) <= nrows;   // block-uniform

    // 32-bit LDS byte offset of smem (low 32 bits of the LDS-aperture address)
    const uint32_t lds_base = (uint32_t)(uintptr_t)(&smem[0]);

    // ---------------- Phase 1: stage noise rows into LDS ----------------
    if (full) {
        const float* gsrc = noise + (size_t)r0 * BLK;   // 16B-aligned region
#pragma unroll
        for (int i = 0; i < NROUNDS; ++i) {
            const uint32_t idx  = (uint32_t)(i * TPB + tid);   // float4 index
            const uint32_t row  = idx >> 4;                    // 16 float4 per row
            const uint32_t k4   = idx & 15u;
            const uint32_t goff = idx * 16u;                   // global byte offset
            const uint32_t lofs = lds_base + row * (ROW_PAD * 4u) + k4 * 16u;
            asm volatile("global_load_async_to_lds_b128 %0, %1, %2"
                         :: "v"(lofs), "v"(goff), "s"(gsrc)
                         : "memory");
        }
        asm volatile("s_wait_asynccnt 0" ::: "memory");
    } else {
        for (int i = tid; i < ROWS_PB * BLK; i += TPB) {
            const int row = i >> 6, k = i & 63;
            float v = 0.0f;
            if (r0 + row < nrows) v = noise[(size_t)(r0 + row) * BLK + k];
            smem[row * ROW_PAD + k] = v;
        }
    }
    __syncthreads();

    // ---------------- Phase 2: per-row compute (one row per thread) -----
    const int row = r0 + tid;
    if (row < nrows) {
        const float4 p = ((const float4*)params)[row];
        const float attack_s  = 0.0005f + p.x * 0.0495f;
        const float energy    = p.y;
        const float tilt      = p.z * 2.0f - 1.0f;
        const float bandwidth = 0.05f + p.w * 0.95f;
        const float attack_samples = fmaxf(attack_s * 16000.0f, 1.0f);
        const float ninv = -1.0f / attack_samples;   // env(k) = exp(k * ninv)

        int d = (int)((float)BLK * (0.5f + 0.5f * bandwidth)); // trunc, like astype(i32)
        d = min(max(d, 1), 480);

        float* myrow = &smem[tid * ROW_PAD];
        float y[BLK];                 // unroll-constant indices -> stays in VGPRs
        float sumsq = 0.0f;
#pragma unroll
        for (int k = 0; k < BLK; ++k) {
            const float env = __expf((float)k * ninv);  // v_exp_f32 path
            const float x   = myrow[k] * env * energy;  // burst sample
            const int   j   = k - d;
            // myrow[j] already holds y[j]: full comb recurrence (any d)
            const float fb  = (j >= 0) ? tilt * myrow[j] : 0.0f;
            const float yk  = x + fb;
            myrow[k] = yk;            // in-place comb state for future taps
            y[k] = yk;
            sumsq += yk * yk;
        }
        const float rinv = 1.0f / sqrtf(sumsq * (1.0f / (float)BLK) + 1e-5f);
#pragma unroll
        for (int k = 0; k < BLK; ++k) myrow[k] = y[k] * rinv;  // store-only pass
    }
    __syncthreads();

    // ---------------- Phase 3: write back via async store from LDS ------
    if (full) {
        float* gdst = out + (size_t)r0 * BLK;
#pragma unroll
        for (int i = 0; i < NROUNDS; ++i) {
            const uint32_t idx  = (uint32_t)(i * TPB + tid);
            const uint32_t row2 = idx >> 4;
            const uint32_t k4   = idx & 15u;
            const uint32_t goff = idx * 16u;
            const uint32_t lofs = lds_base + row2 * (ROW_PAD * 4u) + k4 * 16u;
            asm volatile("global_store_async_from_lds_b128 %0, %1, %2"
                         :: "v"(goff), "v"(lofs), "s"(gdst)
                         : "memory");
        }
        asm volatile("s_wait_asynccnt 0" ::: "memory");
    } else {
        for (int i = tid; i < ROWS_PB * BLK; i += TPB) {
            const int row2 = i >> 6, k = i & 63;
            if (r0 + row2 < nrows)
                out[(size_t)(r0 + row2) * BLK + k] = smem[row2 * ROW_PAD + k];
        }
    }
}

extern "C" void kernel_launch(void* const* d_in, const int* in_sizes, int n_in,
                              void* d_out, int out_size, void* d_ws, size_t ws_size,
                              hipStream_t stream) {
    (void)n_in; (void)out_size; (void)d_ws; (void)ws_size;
    const float* params = (const float*)d_in[0];   // [B,T,4] float32
    const float* noise  = (const float*)d_in[1];   // [B,T,64] float32
    float* out = (float*)d_out;                    // [B, T*64] float32

    const int nrows  = in_sizes[0] / 4;            // B*T
    const int blocks = (nrows + ROWS_PB - 1) / ROWS_PB;

    hipLaunchKernelGGL(transient_comb_kernel, dim3(blocks), dim3(TPB), 0, stream,
                       params, noise, out, nrows);
}